// ROIPoolingLayer_51384988729462
// MI455X (gfx1250) — compile-verified
//
#include <hip/hip_runtime.h>
#include <stdint.h>

// ROI bilinear pooling for MI455X (gfx1250, wave32).
//
// Memory-bound gather (~180MB read / 42MB write => ~10us @ 23.3 TB/s).
// CDNA5 paths used:
//  * global_load_async_to_lds_b128 + s_wait_asynccnt : stage the gathered
//    source pixels (2 rows x 20 cols x 256ch = 40KB per output row) in LDS.
//  * v_wmma_f32_16x16x4_f32 : horizontal resize as a compacted matmul
//    tmp[pw, c] = Wx'[pw, k] * G[k, c], K = 20 (18 gathered cols + 2 zero-
//    weight pads), 5 k-steps of 4. Same HBM traffic as the scalar 2-tap
//    version; matrix flops are free.
//  * vertical pass = uniform 2-tap blend on the accumulator VGPRs.

typedef __attribute__((ext_vector_type(2))) float v2f;
typedef __attribute__((ext_vector_type(8))) float v8f;

#define PHN 9
#define PWN 9
#define CCH 256
#define RPB 128
#define WDIM 256
#define HDIM 256

__global__ __launch_bounds__(256) void roi_pool_wmma_kernel(
    const float* __restrict__ fm, const int* __restrict__ rois,
    float* __restrict__ out)
{
  const int tid   = threadIdx.x;
  const int lane  = tid & 31;
  const int wv    = tid >> 5;          // wave id 0..7
  const int n     = lane & 15;         // N / M index within half-wave
  const int khalf = (lane >> 4) << 1;  // 0 or 2 (A/B K-split across halves)
  const int b = blockIdx.x / RPB;
  const int r = blockIdx.x % RPB;

  // Gathered source pixels: [rowslot(2) * 20 colslots][256 ch]
  __shared__ float sG[2 * 20 * CCH];   // 40 KB
  __shared__ float sfx[PWN];
  __shared__ int   sxa[20];

  const int rb = (b * RPB + r) * 4;
  const int rx = rois[rb + 0], ry = rois[rb + 1];
  const int rw = rois[rb + 2], rh = rois[rb + 3];
  const float wf = (float)rw, hf = (float)rh;

  if (tid < PWN) {
    float sx = ((float)tid + 0.5f) * (wf / (float)PWN) - 0.5f;
    sx = fminf(fmaxf(sx, 0.0f), wf - 1.0f);
    float x0 = floorf(sx);
    int x0i = (int)x0;
    int x1i = min(x0i + 1, rw - 1);
    sfx[tid] = sx - x0;
    sxa[2 * tid + 0] = rx + x0i;
    sxa[2 * tid + 1] = rx + x1i;
  }
  if (tid == 0) {
    // K-pad slots 18/19: zero weight, but must point at valid memory.
    float sx = 0.5f * (wf / (float)PWN) - 0.5f;
    sx = fminf(fmaxf(sx, 0.0f), wf - 1.0f);
    int x0i = (int)floorf(sx);
    sxa[18] = rx + x0i;
    sxa[19] = rx + x0i;
  }
  __syncthreads();

  // A operand (compacted Wx'), per lane, per k-step; independent of ph.
  // A[m,k]: k==2m -> 1-fx[m]; k==2m+1 -> fx[m]; else 0.  (m = pw, rows 9..15 zero)
  // Layout (16x4 f32 A): v0 = K=k0 (lanes 0-15) / K=k0+2 (lanes 16-31); v1 = +1.
  float aop0[5], aop1[5];
  {
    const int m = n;
    #pragma unroll
    for (int ks = 0; ks < 5; ++ks) {
      int k0 = ks * 4 + khalf;
      float a0 = 0.0f, a1 = 0.0f;
      if (m < PWN) {
        if (k0 == 2 * m)          a0 = 1.0f - sfx[m];
        else if (k0 == 2 * m + 1) a0 = sfx[m];
        int k1 = k0 + 1;
        if (k1 == 2 * m)          a1 = 1.0f - sfx[m];
        else if (k1 == 2 * m + 1) a1 = sfx[m];
      }
      aop0[ks] = a0;
      aop1[ks] = a1;
    }
  }

  const float* fmb = fm + (size_t)b * HDIM * WDIM * CCH;
  float* outb = out + (size_t)(b * RPB + r) * PHN * PWN * CCH;
  const uint32_t lds_base = (uint32_t)(uintptr_t)(&sG[0]);

  for (int ph = 0; ph < PHN; ++ph) {
    float sy = ((float)ph + 0.5f) * (hf / (float)PHN) - 0.5f;
    sy = fminf(fmaxf(sy, 0.0f), hf - 1.0f);
    float y0 = floorf(sy);
    int y0i = (int)y0;
    int y1i = min(y0i + 1, rh - 1);
    float fy = sy - y0;
    const int ya0 = ry + y0i, ya1 = ry + y1i;

    __syncthreads();  // previous iteration's consumers of sG are done

    // Async gather: 40 pixel vectors (1KB each) = 80 half-vector (512B)
    // transfers, 10 per wave. Each lane moves 16 bytes (B128).
    #pragma unroll
    for (int i = 0; i < 10; ++i) {
      int o   = wv * 10 + i;   // 0..79
      int p   = o >> 1;        // pixel slot 0..39  (= rowslot*20 + colslot)
      int h2  = o & 1;         // which 512B half of the channel vector
      int rs  = p / 20;
      int cs  = p - rs * 20;
      int yy  = rs ? ya1 : ya0;
      int xx  = sxa[cs];
      uint32_t goff = (uint32_t)((yy * WDIM + xx) * CCH) * 4u
                      + (uint32_t)(h2 * 512) + (uint32_t)(lane * 16);
      uint32_t loff = lds_base + (uint32_t)(p * CCH * 4)
                      + (uint32_t)(h2 * 512) + (uint32_t)(lane * 16);
      asm volatile("global_load_async_to_lds_b128 %0, %1, %2"
                   :: "v"(loff), "v"(goff), "s"(fmb) : "memory");
    }
    asm volatile("s_wait_asynccnt 0x0" ::: "memory");
    __syncthreads();

    const float omfy = 1.0f - fy;
    #pragma unroll
    for (int tt = 0; tt < 2; ++tt) {
      const int tile = wv * 2 + tt;      // 16-channel tile 0..15
      v8f accT = {};                     // top row (rs=0): tmp[pw, c]
      v8f accB = {};                     // bottom row (rs=1)
      #pragma unroll
      for (int ks = 0; ks < 5; ++ks) {
        const int k0 = ks * 4 + khalf;
        v2f a;  a[0] = aop0[ks];  a[1] = aop1[ks];
        // B (4x16 f32): v0 = row K=k0 (lanes 0-15) / K=k0+2 (lanes 16-31); v1 = +1
        v2f bT, bB;
        bT[0] = sG[(0 * 20 + k0    ) * CCH + tile * 16 + n];
        bT[1] = sG[(0 * 20 + k0 + 1) * CCH + tile * 16 + n];
        bB[0] = sG[(1 * 20 + k0    ) * CCH + tile * 16 + n];
        bB[1] = sG[(1 * 20 + k0 + 1) * CCH + tile * 16 + n];
        accT = __builtin_amdgcn_wmma_f32_16x16x4_f32(
            false, a, false, bT, (short)0, accT, false, false);
        accB = __builtin_amdgcn_wmma_f32_16x16x4_f32(
            false, a, false, bB, (short)0, accB, false, false);
      }
      // Vertical 2-tap blend + store.  D layout: VGPR j = pw=j (lanes 0-15),
      // pw=j+8 (lanes 16-31); column = channel-within-tile.
      #pragma unroll
      for (int j = 0; j < 8; ++j) {
        int pw = j + ((lane >> 4) << 3);
        float v = accT[j] * omfy + accB[j] * fy;
        if (pw < PWN) {
          outb[(ph * PWN + pw) * CCH + tile * 16 + n] = v;
        }
      }
    }
  }
}

extern "C" void kernel_launch(void* const* d_in, const int* in_sizes, int n_in,
                              void* d_out, int out_size, void* d_ws, size_t ws_size,
                              hipStream_t stream) {
  const float* fm   = (const float*)d_in[0];   // [4,256,256,256] f32
  const int*   rois = (const int*)d_in[1];     // [4,128,4] i32 (x,y,w,h)
  float*       out  = (float*)d_out;           // [4,128,9,9,256] f32
  (void)in_sizes; (void)n_in; (void)out_size; (void)d_ws; (void)ws_size;
  dim3 grid(4 * RPB);   // one workgroup per ROI
  roi_pool_wmma_kernel<<<grid, 256, 0, stream>>>(fm, rois, out);
}